// ToRGBLayer_59390807769597
// MI455X (gfx1250) — compile-verified
//
#include <hip/hip_runtime.h>

typedef __attribute__((ext_vector_type(2))) float v2f;
typedef __attribute__((ext_vector_type(8))) float v8f;

#define C_DIM 512
#define HW    4096
#define OUT_C 3
#define NPAD  16     // N dimension padded to WMMA width
#define B_STR 522    // LDS row stride (even for b64 align; 522%64=10 decorrelates banks)

// ---------------------------------------------------------------------------
// Kernel 1: styles = w @ (affine*gain)^T + affine_bias ; dcoefs = rsqrt(sum((W*s)^2)+eps)
//           eff[b][o][c] = weight[o][c] * styles[b][c] * dcoefs[b][o]
// One block per batch element. Tiny vs. the 134 MB main stream.
// ---------------------------------------------------------------------------
__global__ __launch_bounds__(256) void torgb_phase1(
    const float* __restrict__ w, const float* __restrict__ aw,
    const float* __restrict__ ab, const float* __restrict__ wt,
    float* __restrict__ eff)
{
    __shared__ float s_w[C_DIM];
    __shared__ float s_styles[C_DIM];
    __shared__ float s_red[OUT_C][256];
    __shared__ float s_d[OUT_C];

    const int b   = blockIdx.x;
    const int tid = threadIdx.x;

    s_w[tid]       = w[b * C_DIM + tid];
    s_w[tid + 256] = w[b * C_DIM + tid + 256];
    __syncthreads();

    const float gain = 0.044194173824159216f; // 1/sqrt(512)
    for (int cc = tid; cc < C_DIM; cc += 256) {
        const float4* row = (const float4*)(aw + (size_t)cc * C_DIM);
        float sum = 0.0f;
        #pragma unroll 4
        for (int k = 0; k < C_DIM / 4; ++k) {
            float4 a = row[k];
            const float* wk = &s_w[k * 4];
            sum += a.x * wk[0] + a.y * wk[1] + a.z * wk[2] + a.w * wk[3];
        }
        s_styles[cc] = sum * gain + ab[cc];
    }
    __syncthreads();

    float p0 = 0.0f, p1 = 0.0f, p2 = 0.0f;
    for (int cc = tid; cc < C_DIM; cc += 256) {
        float s = s_styles[cc];
        float t0 = wt[0 * C_DIM + cc] * s;
        float t1 = wt[1 * C_DIM + cc] * s;
        float t2 = wt[2 * C_DIM + cc] * s;
        p0 += t0 * t0; p1 += t1 * t1; p2 += t2 * t2;
    }
    s_red[0][tid] = p0; s_red[1][tid] = p1; s_red[2][tid] = p2;
    __syncthreads();
    for (int off = 128; off > 0; off >>= 1) {
        if (tid < off) {
            s_red[0][tid] += s_red[0][tid + off];
            s_red[1][tid] += s_red[1][tid + off];
            s_red[2][tid] += s_red[2][tid + off];
        }
        __syncthreads();
    }
    if (tid < OUT_C) s_d[tid] = rsqrtf(s_red[tid][0] + 1e-8f);
    __syncthreads();

    for (int cc = tid; cc < C_DIM; cc += 256) {
        float s = s_styles[cc];
        #pragma unroll
        for (int n = 0; n < OUT_C; ++n)
            eff[((size_t)b * OUT_C + n) * C_DIM + cc] = wt[n * C_DIM + cc] * s * s_d[n];
    }
}

// ---------------------------------------------------------------------------
// Kernel 2: y[b][o][hw] = sum_c eff[b][o][c] * x[b][c][hw] + bias[o]
// WMMA f32 16x16x4: A = 16 pixels x 4 channels (per-lane layout = M row per
// lane, K pair per half-wave), B = 4 channels x 16 outputs from LDS (cols 3..15
// zero). x streamed once with non-temporal loads.
// ---------------------------------------------------------------------------
__global__ __launch_bounds__(256) void torgb_main(
    const float* __restrict__ x, const float* __restrict__ eff,
    const float* __restrict__ bias, float* __restrict__ out)
{
    __shared__ float sB[NPAD * B_STR];

    const int b   = blockIdx.x >> 5;   // 32 blocks per batch image
    const int grp = blockIdx.x & 31;
    const int tid = threadIdx.x;

    // Stage eff[b][0..2][*] into LDS, rows 3..15 (and pad) zeroed. Disjoint writes.
    for (int i = tid; i < NPAD * B_STR; i += 256) {
        int n = i / B_STR;
        int c = i - n * B_STR;
        float v = 0.0f;
        if (n < OUT_C && c < C_DIM) v = eff[((size_t)b * OUT_C + n) * C_DIM + c];
        sB[i] = v;
    }
    __syncthreads();

    const int lane = tid & 31;
    const int wv   = tid >> 5;
    const int m    = lane & 15;        // A: M row / B: N column / D: N column
    const int half = lane >> 4;        // K pair select (K+0,1 vs K+2,3)
    const int koff = half << 1;
    const int hw0  = (grp * 8 + wv) * 16;

    const float* pA = x + ((size_t)b * C_DIM + (size_t)koff) * HW + hw0 + m;
    const float* pB = &sB[m * B_STR + koff];

    v8f acc = {0.f, 0.f, 0.f, 0.f, 0.f, 0.f, 0.f, 0.f};

    #pragma unroll 4
    for (int c = 0; c < C_DIM; c += 4) {
        float a0 = __builtin_nontemporal_load(&pA[(size_t)c * HW]);        // K = c+koff
        float a1 = __builtin_nontemporal_load(&pA[(size_t)(c + 1) * HW]);  // K = c+koff+1
        v2f Bv = *(const v2f*)&pB[c];                                      // ds_load_b64
        v2f Av = {a0, a1};
        acc = __builtin_amdgcn_wmma_f32_16x16x4_f32(
            false, Av, false, Bv, (short)0, acc, false, false);
    }

    if (m < OUT_C) {
        const float bs = bias[m];
        float* po = out + ((size_t)b * OUT_C + m) * HW + hw0 + half * 8;
        #pragma unroll
        for (int r = 0; r < 8; ++r)
            po[r] = acc[r] + bs;       // D: VGPR r holds M = r + half*8
    }
}

extern "C" void kernel_launch(void* const* d_in, const int* in_sizes, int n_in,
                              void* d_out, int out_size, void* d_ws, size_t ws_size,
                              hipStream_t stream) {
    const float* x    = (const float*)d_in[0];  // [16,512,64,64]
    const float* w    = (const float*)d_in[1];  // [16,512]
    const float* aw   = (const float*)d_in[2];  // [512,512]
    const float* ab   = (const float*)d_in[3];  // [512]
    const float* wt   = (const float*)d_in[4];  // [3,512]
    const float* bias = (const float*)d_in[5];  // [3]
    float* out = (float*)d_out;                 // [16,3,64,64]
    float* eff = (float*)d_ws;                  // [16,3,512] = 98 KB

    torgb_phase1<<<16, 256, 0, stream>>>(w, aw, ab, wt, eff);
    torgb_main<<<512, 256, 0, stream>>>(x, eff, bias, out);
}